// SpatialGraphConv_87033217286507
// MI455X (gfx1250) — compile-verified
//
#include <hip/hip_runtime.h>

typedef __attribute__((ext_vector_type(2))) float v2f;
typedef __attribute__((ext_vector_type(4))) float v4f;
typedef __attribute__((ext_vector_type(8))) float v8f;

#define LDA_STRIDE 68   // 64 + 4 pad: spreads LDS banks for the A-fragment reads

// --- Kernel 1: Ahat = D^{-1/2} (A + I) D^{-1/2}  ->  d_ws (64*64 floats) -----
__global__ void gcn_prep_kernel(const float* __restrict__ adj,
                                float* __restrict__ ahat) {
    __shared__ float dinv[64];
    const int tid = threadIdx.x;
    if (tid < 64) {
        float deg = 1.0f;  // self loop
        const float* row = adj + tid * 64;
        #pragma unroll 8
        for (int d = 0; d < 64; ++d) deg += row[d];
        dinv[tid] = rsqrtf(deg);
    }
    __syncthreads();
    for (int i = tid; i < 64 * 64; i += 256) {
        const int c = i >> 6, d = i & 63;
        const float a = adj[i] + (c == d ? 1.0f : 0.0f);
        ahat[i] = a * dinv[c] * dinv[d];
    }
}

// --- Kernel 2: agg = Ahat @ x[b] via WMMA f32 16x16x4, then expand by W,b ----
// Grid: 1024 blocks x 128 threads (4 waves). Wave task = (batch, 16-col t-tile).
__global__ void __launch_bounds__(128) gcn_wmma_kernel(
    const float* __restrict__ x,     // [128,64,512]
    const float* __restrict__ ahat,  // [64,64]
    const float* __restrict__ W,     // [16]
    const float* __restrict__ bias,  // [16]
    float* __restrict__ out)         // [128,64,16,512]
{
    __shared__ __align__(16) float ldsA[64 * LDA_STRIDE]; // padded Ahat
    __shared__ __align__(16) float ldsT[4 * 1024];        // per-wave 64x16 agg
    __shared__ float ldsW[16];
    __shared__ float ldsB[16];

    const int tid = threadIdx.x;

    // Stage Ahat into LDS (1024 v4f / 128 threads = 8 each)
    const v4f* src4 = reinterpret_cast<const v4f*>(ahat);
    v4f* A4 = reinterpret_cast<v4f*>(ldsA);
    #pragma unroll
    for (int j = 0; j < 8; ++j) {
        const int i = tid + 128 * j;      // v4f index 0..1023
        const int c = i >> 4, dq = i & 15;
        A4[c * (LDA_STRIDE / 4) + dq] = src4[i];
    }
    if (tid < 16)      ldsW[tid] = W[tid];
    else if (tid < 32) ldsB[tid - 16] = bias[tid - 16];
    __syncthreads();

    const int lane = tid & 31;
    const int wid  = tid >> 5;
    const int task  = blockIdx.x * 4 + wid;   // 0..4095
    const int b     = task >> 5;              // batch 0..127
    const int tbase = (task & 31) << 4;       // t-tile base 0..496

    const int tcol = lane & 15;  // A row-in-tile / B column
    const int half = lane >> 4;  // K half select per ISA fragment layout

    v8f acc[4] = {};  // 4 M-tiles (c = 0..63) of the 16x16 f32 accumulator

    const float* xb = x + (size_t)b * (64 * 512) + tbase + tcol;

    #pragma unroll
    for (int kk = 0; kk < 16; ++kk) {         // K = 64 in steps of 4
        const int k0 = kk * 4 + half * 2;
        v2f bfrag;
        bfrag[0] = __builtin_nontemporal_load(&xb[(size_t)k0 * 512]);
        bfrag[1] = __builtin_nontemporal_load(&xb[(size_t)(k0 + 1) * 512]);
        #pragma unroll
        for (int m = 0; m < 4; ++m) {         // reuse B across 4 M-tiles
            const float* ar = &ldsA[(m * 16 + tcol) * LDA_STRIDE + k0];
            v2f afrag;
            afrag[0] = ar[0];
            afrag[1] = ar[1];
            acc[m] = __builtin_amdgcn_wmma_f32_16x16x4_f32(
                false, afrag, false, bfrag, (short)0, acc[m], false, false);
        }
    }

    // Stage all 4 accumulator tiles to wave-private LDS.
    // C/D layout: vgpr v, lane l -> row (v + 8*(l>>4)), col (l&15).
    float* tileBase = &ldsT[wid * 1024];
    #pragma unroll
    for (int m = 0; m < 4; ++m)
        #pragma unroll
        for (int v = 0; v < 8; ++v)
            tileBase[m * 256 + (v + 8 * half) * 16 + tcol] = acc[m][v];
    __syncthreads();  // uniform across all 4 waves

    // Expand: out[b, m*16+row, o, tbase+4q..] = agg * W[o] + bias[o]
    // 4096 v4f chunks / 32 lanes = 128 non-temporal B128 stores per lane.
    float* outBase = out + (size_t)b * (64 * 16 * 512) + tbase;
    const v4f* t4 = reinterpret_cast<const v4f*>(tileBase);
    #pragma unroll 4
    for (int i = 0; i < 128; ++i) {
        const int cid = lane + 32 * i;        // 0..4095
        const int m   = cid >> 10;
        const int row = (cid >> 6) & 15;
        const int o   = (cid >> 2) & 15;
        const int q   = cid & 3;
        const float w  = ldsW[o];
        const float bb = ldsB[o];
        const v4f t = t4[m * 64 + row * 4 + q];
        v4f r;
        r[0] = fmaf(t[0], w, bb);
        r[1] = fmaf(t[1], w, bb);
        r[2] = fmaf(t[2], w, bb);
        r[3] = fmaf(t[3], w, bb);
        v4f* dst = reinterpret_cast<v4f*>(
            outBase + ((size_t)((m * 16 + row) * 16 + o) << 9) + 4 * q);
        __builtin_nontemporal_store(r, dst);
    }
}

extern "C" void kernel_launch(void* const* d_in, const int* in_sizes, int n_in,
                              void* d_out, int out_size, void* d_ws, size_t ws_size,
                              hipStream_t stream) {
    const float* x    = (const float*)d_in[0];  // [128,64,512]
    const float* adj  = (const float*)d_in[1];  // [64,64]
    const float* W    = (const float*)d_in[2];  // [1,16] flat
    const float* bias = (const float*)d_in[3];  // [16]
    float* out  = (float*)d_out;                // [128,64,16,512]
    float* ahat = (float*)d_ws;                 // 64*64 floats scratch

    gcn_prep_kernel<<<1, 256, 0, stream>>>(adj, ahat);
    gcn_wmma_kernel<<<1024, 128, 0, stream>>>(x, ahat, W, bias, out);
}